// GraphClassification_51539607552910
// MI455X (gfx1250) — compile-verified
//
#include <hip/hip_runtime.h>
#include <hip/hip_bf16.h>

typedef __attribute__((ext_vector_type(2))) float v2f;
typedef __attribute__((ext_vector_type(8))) float v8f;

// Problem constants (from reference)
#define B_   16
#define L_   512
#define DB_  3072
#define D_   300
#define NN_  256
#define NW_  30
#define H_   256

#define M1   (B_ * L_)      // 8192 rows of big GEMM
#define MN   (B_ * NN_)     // 4096 nodes

// ---------------- Kernel 1: art = PReLU(A @ W^T + b) -----------------------
// A: [8192,3072], W: [300,3072] row-major. Out: [8192,300].
// Block: 128 thr (4 waves). Tile: BM=128 x BN=64, K-chunk KC=32.
// Both As and Bs stored row-contiguous-in-K ([m][k] / [n][k], stride 36):
// fragment x/y are adjacent in LDS -> ds_load_b64 into aligned VGPR pairs,
// and staging is a straight float4 copy (W rows are contiguous in k).
#define BM 128
#define BN 64
#define KC 32
#define LST 36   // LDS row stride (32+4): float4-aligned, conflict-free frag reads

__global__ __launch_bounds__(128) void gemm_bert(
    const float* __restrict__ A, const float* __restrict__ W,
    const float* __restrict__ bias, const float* __restrict__ prelu_a,
    float* __restrict__ art)
{
    __shared__ __align__(16) float As[BM][LST];
    __shared__ __align__(16) float Bs[BN][LST];

    const int tid  = threadIdx.x;
    const int wave = tid >> 5;
    const int lane = tid & 31;
    const int rc   = lane & 15;          // row (A) / col (B,C)
    const int rsel = (lane >> 4) << 1;   // K-pair select: 0 or 2
    const int rowh = (lane >> 4) << 3;   // C-row half offset: 0 or 8

    const int m0 = blockIdx.x * BM;      // gridDim.x = 64
    const int nb = blockIdx.y * BN;      // gridDim.y = 5 (covers 300, zero-padded)

    v8f acc[8];                          // [nt] rows wave*32, [nt+4] rows wave*32+16
#pragma unroll
    for (int t = 0; t < 8; ++t) acc[t] = (v8f){0,0,0,0,0,0,0,0};

    for (int kb = 0; kb < DB_; kb += KC) {
        // Stage A tile (128x32): 1024 float4, 8 per thread
#pragma unroll
        for (int j = 0; j < 8; ++j) {
            int i   = tid + 128 * j;
            int row = i >> 3;            // 8 float4 per row
            int c4  = (i & 7) << 2;
            float4 v = *(const float4*)&A[(size_t)(m0 + row) * DB_ + kb + c4];
            *(float4*)&As[row][c4] = v;
        }
        // Stage B tile (64 rows of W x 32): 512 float4, 4 per thread; zero-pad n>=300
#pragma unroll
        for (int j = 0; j < 4; ++j) {
            int i  = tid + 128 * j;
            int n  = i >> 3;
            int c4 = (i & 7) << 2;
            float4 v = make_float4(0.f, 0.f, 0.f, 0.f);
            if (nb + n < D_)
                v = *(const float4*)&W[(size_t)(nb + n) * DB_ + kb + c4];
            *(float4*)&Bs[n][c4] = v;
        }
        __syncthreads();

        const int ar0 = wave * 32 + rc;       // wave's 32-row strip
        const int ar1 = wave * 32 + 16 + rc;
#pragma unroll
        for (int kk = 0; kk < KC; kk += 4) {
            v2f a0, a1;
            a0.x = As[ar0][kk + rsel];
            a0.y = As[ar0][kk + rsel + 1];
            a1.x = As[ar1][kk + rsel];
            a1.y = As[ar1][kk + rsel + 1];
#pragma unroll
            for (int nt = 0; nt < 4; ++nt) {
                v2f b;
                b.x = Bs[nt * 16 + rc][kk + rsel];
                b.y = Bs[nt * 16 + rc][kk + rsel + 1];
                acc[nt] = __builtin_amdgcn_wmma_f32_16x16x4_f32(
                    false, a0, false, b, (short)0, acc[nt], false, false);
                acc[nt + 4] = __builtin_amdgcn_wmma_f32_16x16x4_f32(
                    false, a1, false, b, (short)0, acc[nt + 4], false, false);
            }
        }
        __syncthreads();
    }

    const float pa = prelu_a[0];
#pragma unroll
    for (int nt = 0; nt < 4; ++nt) {
        int gc = nb + nt * 16 + rc;
        if (gc < D_) {
            float bb = bias[gc];
#pragma unroll
            for (int part = 0; part < 2; ++part) {
#pragma unroll
                for (int i = 0; i < 8; ++i) {
                    int gr = m0 + wave * 32 + part * 16 + i + rowh;
                    float v = acc[nt + 4 * part][i] + bb;
                    v = v > 0.f ? v : pa * v;
                    art[(size_t)gr * D_ + gc] = v;
                }
            }
        }
    }
}

// ---------------- Kernel 2: node gather + masked mean pool ------------------
__global__ __launch_bounds__(128) void gather_pool(
    const float* __restrict__ art, const int* __restrict__ idx,
    const int* __restrict__ mask, float* __restrict__ nodes)
{
    const int bn = blockIdx.x;           // 0..4095
    const int b  = bn / NN_;
    __shared__ int sidx[NW_];
    __shared__ int smsk[NW_];
    if (threadIdx.x < NW_) {
        sidx[threadIdx.x] = idx[(size_t)bn * NW_ + threadIdx.x];
        smsk[threadIdx.x] = mask[(size_t)bn * NW_ + threadIdx.x];
    }
    __syncthreads();

    float cnt = 0.f;
#pragma unroll
    for (int w = 0; w < NW_; ++w) cnt += (float)smsk[w];
    const float inv = 1.f / fmaxf(cnt, 1.f);

    for (int d = threadIdx.x; d < D_; d += 128) {
        float s = 0.f;
#pragma unroll
        for (int w = 0; w < NW_; ++w) {
            if (smsk[w])
                s += art[((size_t)b * L_ + sidx[w]) * D_ + d];
        }
        nodes[(size_t)bn * D_ + d] = s * inv;
    }
}

// ---------------- Kernel 3: h = relu(nodes @ W1^T + b1), WMMA ---------------
// nodes: [4096,300], W1: [256,300] -> h: [4096,256]. Wave per 16x16 tile.
__global__ __launch_bounds__(128) void gemm_head(
    const float* __restrict__ Nd, const float* __restrict__ W1,
    const float* __restrict__ b1, float* __restrict__ h)
{
    const int wave = threadIdx.x >> 5;
    const int lane = threadIdx.x & 31;
    const int rc   = lane & 15;
    const int rsel = (lane >> 4) << 1;
    const int rowh = (lane >> 4) << 3;

    const int tile = blockIdx.x * 4 + wave;   // gridDim.x = 1024 -> 4096 tiles
    const int mt = tile >> 4;                 // 0..255
    const int nt = tile & 15;                 // 0..15

    const float* Ar = Nd + (size_t)(mt * 16 + rc) * D_; // A row for this lane
    const float* Br = W1 + (size_t)(nt * 16 + rc) * D_; // B[k][n]=W1[n][k]

    v8f acc = (v8f){0,0,0,0,0,0,0,0};
#pragma unroll 5
    for (int k = 0; k < D_; k += 4) {
        v2f a, b;
        a.x = Ar[k + rsel];
        a.y = Ar[k + rsel + 1];
        b.x = Br[k + rsel];
        b.y = Br[k + rsel + 1];
        acc = __builtin_amdgcn_wmma_f32_16x16x4_f32(
            false, a, false, b, (short)0, acc, false, false);
    }

    const int col = nt * 16 + rc;
    const float bb = b1[col];
#pragma unroll
    for (int i = 0; i < 8; ++i) {
        int row = mt * 16 + i + rowh;
        float v = acc[i] + bb;
        h[(size_t)row * H_ + col] = fmaxf(v, 0.f);
    }
}

// ---------------- Kernel 4: out = h . W2 + b2 (wave reduction) --------------
__global__ __launch_bounds__(256) void head_out(
    const float* __restrict__ h, const float* __restrict__ W2,
    const float* __restrict__ b2, float* __restrict__ out)
{
    const int j    = blockIdx.x * 8 + (threadIdx.x >> 5); // gridDim.x = 512
    const int lane = threadIdx.x & 31;
    const float* hr = h + (size_t)j * H_;
    float p = 0.f;
#pragma unroll
    for (int e = lane; e < H_; e += 32) p += hr[e] * W2[e];
#pragma unroll
    for (int off = 16; off > 0; off >>= 1) p += __shfl_down(p, off, 32);
    if (lane == 0) out[j] = p + b2[0];
}

// ---------------- Launch ----------------------------------------------------
extern "C" void kernel_launch(void* const* d_in, const int* in_sizes, int n_in,
                              void* d_out, int out_size, void* d_ws, size_t ws_size,
                              hipStream_t stream)
{
    const float* articles = (const float*)d_in[0];
    const int*   nidx     = (const int*)  d_in[1];
    const int*   nmask    = (const int*)  d_in[2];
    // d_in[3], d_in[4]: relations (dead code in reference -> skipped)
    const float* W_bert   = (const float*)d_in[5];
    const float* b_bert   = (const float*)d_in[6];
    const float* prelu_a  = (const float*)d_in[7];
    const float* W1       = (const float*)d_in[8];
    const float* b1       = (const float*)d_in[9];
    const float* W2       = (const float*)d_in[10];
    const float* b2       = (const float*)d_in[11];
    float* out = (float*)d_out;

    float* art   = (float*)d_ws;                 // 8192*300 floats
    float* nodes = art + (size_t)M1 * D_;        // 4096*300 floats
    float* h     = nodes + (size_t)MN * D_;      // 4096*256 floats
    (void)in_sizes; (void)n_in; (void)out_size; (void)ws_size;

    gemm_bert  <<<dim3(M1 / BM, 5), 128, 0, stream>>>(articles, W_bert, b_bert, prelu_a, art);
    gather_pool<<<MN, 128, 0, stream>>>(art, nidx, nmask, nodes);
    gemm_head  <<<MN * (H_ / 16) / (4 * 16), 128, 0, stream>>>(nodes, W1, b1, h);
    head_out   <<<MN / 8, 256, 0, stream>>>(h, W2, b2, out);
}